// impactDetect_64862596104601
// MI455X (gfx1250) — compile-verified
//
#include <hip/hip_runtime.h>
#include <hip/hip_bf16.h>

typedef __attribute__((ext_vector_type(2))) float v2f;
typedef __attribute__((ext_vector_type(8))) float v8f;

#define HD 128  // hidden dim == in dim == 128

// ---------------------------------------------------------------------------
// degree / norm kernels
// ---------------------------------------------------------------------------
__global__ void zero_int_kernel(int* __restrict__ p, int n) {
    int i = blockIdx.x * blockDim.x + threadIdx.x;
    if (i < n) p[i] = 0;
}

__global__ void deg_count_kernel(const int* __restrict__ dst, int* __restrict__ cnt, int nE) {
    int e = blockIdx.x * blockDim.x + threadIdx.x;
    if (e < nE) atomicAdd(&cnt[dst[e]], 1);
}

__global__ void dinv_kernel(const int* __restrict__ cnt, float* __restrict__ dinv, int n) {
    int i = blockIdx.x * blockDim.x + threadIdx.x;
    if (i < n) dinv[i] = rsqrtf((float)cnt[i] + 1.0f);  // +1 self loop, deg>=1
}

// ---------------------------------------------------------------------------
// GEMM: out[nrows x 128] = A[nrows x 128] @ W[128 x 128]  (row-major)
// One wave per 16x16 output tile, f32 WMMA 16x16x4, K=128 -> 32 wmma ops.
// EPI: 0 = none, 1 = bias + leaky_relu(0.01)
// ---------------------------------------------------------------------------
template <int EPI>
__global__ void gemm128_wmma_kernel(const float* __restrict__ A,
                                    const float* __restrict__ W,
                                    const float* __restrict__ bias,
                                    float* __restrict__ out, int nrows) {
    const int lane  = threadIdx.x & 31;
    const int wave  = (blockIdx.x * blockDim.x + threadIdx.x) >> 5;
    const int tileM = wave >> 3;        // 8 column tiles (128/16)
    const int tileN = wave & 7;
    if (tileM * 16 >= nrows) return;    // wave-uniform: EXEC stays all-ones

    const int mrow  = lane & 15;        // A: M index, B/D: N index
    const int khalf = (lane >> 4) << 1; // 0 or 2

    const float* Arow = A + (size_t)(tileM * 16 + mrow) * HD;
    const float* Wcol = W + tileN * 16 + mrow;

    v8f c = {};
#pragma unroll
    for (int kb = 0; kb < 32; ++kb) {
        const int k0 = kb * 4 + khalf;
        v2f a, b;
        const float2 av = *(const float2*)(Arow + k0);   // k0 even -> 8B aligned
        a.x = av.x; a.y = av.y;
        b.x = Wcol[(size_t)k0 * HD];
        b.y = Wcol[(size_t)(k0 + 1) * HD];
        c = __builtin_amdgcn_wmma_f32_16x16x4_f32(
                /*neg_a=*/false, a, /*neg_b=*/false, b,
                /*c_mod=*/(short)0, c, /*reuse_a=*/false, /*reuse_b=*/false);
    }

    const int ncol  = tileN * 16 + mrow;
    const int mbase = tileM * 16 + ((lane >> 4) << 3);  // lanes 16-31 -> M+8
#pragma unroll
    for (int r = 0; r < 8; ++r) {
        float v = c[r];
        if (EPI == 1) {
            v += bias[ncol];
            v = v > 0.0f ? v : 0.01f * v;
        }
        out[(size_t)(mbase + r) * HD + ncol] = v;
    }
}

// ---------------------------------------------------------------------------
// accumulator init with self-loop term:  out = h * dinv^2 + bias
// ---------------------------------------------------------------------------
__global__ void selfloop_init_kernel(const float* __restrict__ h,
                                     const float* __restrict__ dinv,
                                     const float* __restrict__ bias,
                                     float* __restrict__ out, int n) {
    int tid = blockIdx.x * blockDim.x + threadIdx.x;
    if (tid >= n * HD) return;
    int i = tid >> 7, f = tid & (HD - 1);
    float di = dinv[i];
    out[tid] = h[tid] * di * di + bias[f];
}

// ---------------------------------------------------------------------------
// edge scatter: out[dst] += h[src] * dinv[src]*dinv[dst].  One wave per edge,
// each lane moves a float4 (32 lanes x 4 = 128 features), atomic f32 adds.
// ---------------------------------------------------------------------------
__global__ void scatter_kernel(const float* __restrict__ h,
                               const int* __restrict__ src,
                               const int* __restrict__ dst,
                               const float* __restrict__ dinv,
                               float* __restrict__ out, int nE) {
    long long tid = (long long)blockIdx.x * blockDim.x + threadIdx.x;
    long long e = tid >> 5;
    if (e >= nE) return;
    int lane = (int)(tid & 31);
    int s = src[e], d = dst[e];
    float norm = dinv[s] * dinv[d];
    float4 v = ((const float4*)(h + (size_t)s * HD))[lane];
    float* op = out + (size_t)d * HD + lane * 4;
    atomicAdd(op + 0, v.x * norm);
    atomicAdd(op + 1, v.y * norm);
    atomicAdd(op + 2, v.z * norm);
    atomicAdd(op + 3, v.w * norm);
}

__global__ void relu_inplace_kernel(float* __restrict__ p, int n) {
    int i = blockIdx.x * blockDim.x + threadIdx.x;
    if (i < n) p[i] = fmaxf(p[i], 0.0f);
}

// ---------------------------------------------------------------------------
// head: out[t] = leaky_relu( Z[idx[t]] . Wy + by ).  One wave per index.
// ---------------------------------------------------------------------------
__global__ void head_dot_kernel(const float* __restrict__ Z,
                                const int* __restrict__ idx,
                                const float* __restrict__ Wy,
                                const float* __restrict__ by,
                                float* __restrict__ out, int T) {
    int tid = blockIdx.x * blockDim.x + threadIdx.x;
    int t = tid >> 5, lane = tid & 31;
    if (t >= T) return;
    int node = idx[t];
    float4 z = ((const float4*)(Z + (size_t)node * HD))[lane];
    float4 w = ((const float4*)Wy)[lane];
    float s = z.x * w.x + z.y * w.y + z.z * w.z + z.w * w.w;
#pragma unroll
    for (int off = 16; off > 0; off >>= 1) s += __shfl_xor(s, off, 32);
    if (lane == 0) {
        float v = s + by[0];
        out[t] = v > 0.0f ? v : 0.01f * v;
    }
}

// ---------------------------------------------------------------------------
// 128 -> 2 projection: out[i][c] = (lrelu?)( Z[i] . W[:,c] + b[c] ), c=0,1.
// W row-major [128 x 2]. One wave per node.
// ---------------------------------------------------------------------------
template <int LRELU>
__global__ void proj2_kernel(const float* __restrict__ Z,
                             const float* __restrict__ W,
                             const float* __restrict__ b,
                             float* __restrict__ out, int n) {
    long long tid = (long long)blockIdx.x * blockDim.x + threadIdx.x;
    int i = (int)(tid >> 5), lane = (int)(tid & 31);
    if (i >= n) return;
    float4 z = ((const float4*)(Z + (size_t)i * HD))[lane];
    int k0 = lane * 4;
    float s0 = z.x * W[(k0 + 0) * 2] + z.y * W[(k0 + 1) * 2] +
               z.z * W[(k0 + 2) * 2] + z.w * W[(k0 + 3) * 2];
    float s1 = z.x * W[(k0 + 0) * 2 + 1] + z.y * W[(k0 + 1) * 2 + 1] +
               z.z * W[(k0 + 2) * 2 + 1] + z.w * W[(k0 + 3) * 2 + 1];
#pragma unroll
    for (int off = 16; off > 0; off >>= 1) {
        s0 += __shfl_xor(s0, off, 32);
        s1 += __shfl_xor(s1, off, 32);
    }
    if (lane == 0) {
        float v0 = s0 + b[0], v1 = s1 + b[1];
        if (LRELU) {
            v0 = v0 > 0.0f ? v0 : 0.01f * v0;
            v1 = v1 > 0.0f ? v1 : 0.01f * v1;
        }
        out[(size_t)i * 2 + 0] = v0;
        out[(size_t)i * 2 + 1] = v1;
    }
}

// ---------------------------------------------------------------------------
template <int EPI>
static inline void launch_gemm(const float* A, const float* W, const float* bias,
                               float* out, int nrows, hipStream_t s) {
    int tilesM = (nrows + 15) / 16;
    long long waves = (long long)tilesM * 8;
    int blocks = (int)((waves * 32 + 127) / 128);
    gemm128_wmma_kernel<EPI><<<blocks, 128, 0, s>>>(A, W, bias, out, nrows);
}

extern "C" void kernel_launch(void* const* d_in, const int* in_sizes, int n_in,
                              void* d_out, int out_size, void* d_ws, size_t ws_size,
                              hipStream_t stream) {
    (void)n_in; (void)out_size; (void)ws_size;
    const float* x    = (const float*)d_in[0];
    const int*   ei   = (const int*)d_in[1];
    const float* fx   = (const float*)d_in[2];
    const int*   fei  = (const int*)d_in[3];
    const int*   tidx = (const int*)d_in[4];
    const int*   cidx = (const int*)d_in[5];
    const float* W1  = (const float*)d_in[6];   const float* b1  = (const float*)d_in[7];
    const float* W2  = (const float*)d_in[8];   const float* b2  = (const float*)d_in[9];
    const float* Wy1 = (const float*)d_in[10];  const float* by1 = (const float*)d_in[11];
    const float* Wy0 = (const float*)d_in[12];  const float* by0 = (const float*)d_in[13];
    const float* Wb  = (const float*)d_in[14];  const float* bb  = (const float*)d_in[15];
    const float* Wp1 = (const float*)d_in[16];  const float* bp1 = (const float*)d_in[17];
    const float* Wp2 = (const float*)d_in[18];  const float* bp2 = (const float*)d_in[19];

    const int N = in_sizes[0] / HD;
    const int E = in_sizes[1] / 2;
    const int T = in_sizes[4];

    float* out = (float*)d_out;
    float* y1      = out;
    float* yc0     = out + T;
    float* y0      = out + 2 * (size_t)T;
    float* yc1     = out + 3 * (size_t)T;
    float* fprob   = out + 4 * (size_t)T;
    float* fprob_f = fprob + 2 * (size_t)N;
    float* tprob   = fprob_f + 2 * (size_t)N;
    float* tprob_f = tprob + 2 * (size_t)N;

    // workspace: two N x 128 buffers + dinv + degree counts (~103 MB)
    float* bufA   = (float*)d_ws;
    float* bufB   = bufA + (size_t)N * HD;
    float* dinv   = bufB + (size_t)N * HD;
    int*   degcnt = (int*)(dinv + N);

    const int BT = 256;
    const int gN   = (N + BT - 1) / BT;
    const int gE   = (E + BT - 1) / BT;
    const int gNF  = (int)(((long long)N * HD + BT - 1) / BT);
    const int gE32 = (int)(((long long)E * 32 + BT - 1) / BT);
    const int gN32 = (int)(((long long)N * 32 + BT - 1) / BT);
    const int gT32 = (T * 32 + BT - 1) / BT;

    for (int g = 0; g < 2; ++g) {
        const float* X   = (g == 0) ? x : fx;
        const int*   src = (g == 0) ? ei : fei;
        const int*   dst = src + E;

        // symmetric-norm degree terms
        zero_int_kernel<<<gN, BT, 0, stream>>>(degcnt, N);
        deg_count_kernel<<<gE, BT, 0, stream>>>(dst, degcnt, E);
        dinv_kernel<<<gN, BT, 0, stream>>>(degcnt, dinv, N);

        // layer 1: h = X @ W1 ; Z1 = relu(scatter(h*norm) + h*dinv^2 + b1)
        launch_gemm<0>(X, W1, nullptr, bufA, N, stream);
        selfloop_init_kernel<<<gNF, BT, 0, stream>>>(bufA, dinv, b1, bufB, N);
        scatter_kernel<<<gE32, BT, 0, stream>>>(bufA, src, dst, dinv, bufB, E);
        relu_inplace_kernel<<<gNF, BT, 0, stream>>>(bufB, N * HD);

        // layer 2: h2 = Z1 @ W2 ; Z2 = scatter(h2*norm) + h2*dinv^2 + b2
        launch_gemm<0>(bufB, W2, nullptr, bufA, N, stream);
        selfloop_init_kernel<<<gNF, BT, 0, stream>>>(bufA, dinv, b2, bufB, N);
        scatter_kernel<<<gE32, BT, 0, stream>>>(bufA, src, dst, dinv, bufB, E);
        // bufB = Z2 (real or fake)

        if (g == 0) {
            head_dot_kernel<<<gT32, BT, 0, stream>>>(bufB, tidx, Wy1, by1, y1, T);
            head_dot_kernel<<<gT32, BT, 0, stream>>>(bufB, cidx, Wy0, by0, y0, T);
            proj2_kernel<0><<<gN32, BT, 0, stream>>>(bufB, Wb, bb, fprob, N);
        } else {
            head_dot_kernel<<<gT32, BT, 0, stream>>>(bufB, tidx, Wy0, by0, yc0, T);
            head_dot_kernel<<<gT32, BT, 0, stream>>>(bufB, cidx, Wy1, by1, yc1, T);
            proj2_kernel<0><<<gN32, BT, 0, stream>>>(bufB, Wb, bb, fprob_f, N);
        }

        // treatment head: P1 = lrelu(Z2 @ Wp1 + bp1) ; tprob = lrelu(P1 @ Wp2 + bp2)
        launch_gemm<1>(bufB, Wp1, bp1, bufA, N, stream);
        proj2_kernel<1><<<gN32, BT, 0, stream>>>(bufA, Wp2, bp2,
                                                 (g == 0) ? tprob : tprob_f, N);
    }
}